// Net_25829933318546
// MI455X (gfx1250) — compile-verified
//
#include <hip/hip_runtime.h>

// ---------------------------------------------------------------------------
// 3-layer GCN for MI455X (gfx1250, wave32).
//   deg/dinv -> norm[e] precompute -> per layer:
//   [WMMA f32 GEMM X*W] -> [edge gather/scatter-add (float4)] ->
//   [finalize: + dinv^2*H (self loop) + bias (+ReLU), float4]
// GEMM uses V_WMMA_F32_16X16X4_F32 (full f32 precision; GEMM is <1% of time).
// Aggregation is the bandwidth phase; per-layer working set (~65 MB incl.
// norm) fits in the 192 MB L2, so gathers + f32 atomics resolve in L2.
// ---------------------------------------------------------------------------

typedef __attribute__((ext_vector_type(2))) float v2f;
typedef __attribute__((ext_vector_type(8))) float v8f;

#define FEAT 64

// ---------------- utility fill ----------------
__global__ void fill_kernel(float* __restrict__ p, float v, long n) {
    long stride = (long)gridDim.x * blockDim.x;
    for (long t = (long)blockIdx.x * blockDim.x + threadIdx.x; t < n; t += stride)
        p[t] = v;
}

// ---------------- degree (in-degree over dst; deg pre-initialized to 1.0 for self loop)
__global__ void deg_kernel(const int* __restrict__ dst, float* __restrict__ deg, long E) {
    long t = (long)blockIdx.x * blockDim.x + threadIdx.x;
    if (t >= E) return;
    unsafeAtomicAdd(deg + dst[t], 1.0f);
}

// ---------------- dinv = rsqrt(deg), in place (deg >= 1 always) --------------
__global__ void dinv_kernel(float* __restrict__ deg, long n) {
    long t = (long)blockIdx.x * blockDim.x + threadIdx.x;
    if (t >= n) return;
    deg[t] = __frsqrt_rn(deg[t]);
}

// ---------------- norm[e] = dinv[src] * dinv[dst]  (reused by all 3 layers) -
__global__ void norm_kernel(const int* __restrict__ src, const int* __restrict__ dst,
                            const float* __restrict__ dinv, float* __restrict__ norm,
                            long E) {
    long t = (long)blockIdx.x * blockDim.x + threadIdx.x;
    if (t >= E) return;
    norm[t] = dinv[src[t]] * dinv[dst[t]];
}

// ---------------- H = X * W  via V_WMMA_F32_16X16X4_F32 ---------------------
// One wave per 16-row M tile, 4 N-tiles (64 cols) as 4 v8f accumulators.
// A layout (16x4 f32): reg0 = X[m][k0 + 2*kh], reg1 = X[m][k0 + 2*kh + 1],
//   m = lane&15, kh = lane>>4.  B mirrors: reg0/1 = W[k][n], W[k+1][n].
// C/D layout: VGPR r -> row r (lanes 0-15) / row r+8 (lanes 16-31), col = lane&15.
__global__ void gemm_wmma_kernel(const float* __restrict__ X,
                                 const float* __restrict__ W,
                                 float* __restrict__ H,
                                 int N, int K) {
    const int lane  = threadIdx.x & 31;
    const int wv    = threadIdx.x >> 5;
    const int mtile = blockIdx.x * (blockDim.x >> 5) + wv;
    const int Mtiles = N >> 4;                    // N is a multiple of 16
    if (mtile >= Mtiles) return;                  // wave-uniform: EXEC stays all-1s

    const int ml = lane & 15;
    const int kh = lane >> 4;                     // 0 or 1

    const float* __restrict__ xrow = X + (long)(mtile * 16 + ml) * K;

    v8f acc0 = {}, acc1 = {}, acc2 = {}, acc3 = {};

    for (int k0 = 0; k0 < K; k0 += 4) {
        const int k = k0 + 2 * kh;
        // A fragment: two consecutive K elements of this lane's row
        const float2 a2 = *reinterpret_cast<const float2*>(xrow + k);
        v2f a; a.x = a2.x; a.y = a2.y;
        // B fragments: rows k and k+1 of W (row-major [K][64]); tiny, L2-resident
        const float* __restrict__ w0 = W + (long)k * FEAT;
        const float* __restrict__ w1 = w0 + FEAT;
        v2f b0; b0.x = w0[ml +  0]; b0.y = w1[ml +  0];
        v2f b1; b1.x = w0[ml + 16]; b1.y = w1[ml + 16];
        v2f b2; b2.x = w0[ml + 32]; b2.y = w1[ml + 32];
        v2f b3; b3.x = w0[ml + 48]; b3.y = w1[ml + 48];

        acc0 = __builtin_amdgcn_wmma_f32_16x16x4_f32(false, a, false, b0, (short)0, acc0, false, false);
        acc1 = __builtin_amdgcn_wmma_f32_16x16x4_f32(false, a, false, b1, (short)0, acc1, false, false);
        acc2 = __builtin_amdgcn_wmma_f32_16x16x4_f32(false, a, false, b2, (short)0, acc2, false, false);
        acc3 = __builtin_amdgcn_wmma_f32_16x16x4_f32(false, a, false, b3, (short)0, acc3, false, false);
    }

    // D: VGPR r holds row r (lanes 0-15) / row r+8 (lanes 16-31)
    float* __restrict__ hrow = H + (long)(mtile * 16 + 8 * kh) * FEAT;
#pragma unroll
    for (int r = 0; r < 8; ++r) {
        hrow[(long)r * FEAT + ml +  0] = acc0[r];
        hrow[(long)r * FEAT + ml + 16] = acc1[r];
        hrow[(long)r * FEAT + ml + 32] = acc2[r];
        hrow[(long)r * FEAT + ml + 48] = acc3[r];
    }
}

// ---------------- edge aggregation: agg[dst] += norm[e] * H[src] ------------
// 16 threads per edge, 4 features each via float4: coalesced 256B gather per
// edge, 4 f32 atomics per thread (L2-resident destination).
__global__ void agg_kernel(const int* __restrict__ src, const int* __restrict__ dst,
                           const float* __restrict__ norm, const float* __restrict__ H,
                           float* __restrict__ agg, long E) {
    long t = (long)blockIdx.x * blockDim.x + threadIdx.x;
    long e = t >> 4;
    if (e >= E) return;
    int f4 = (int)(t & 15) * 4;
    int s = src[e];
    int d = dst[e];
    float w = norm[e];
    const float4 h = *reinterpret_cast<const float4*>(H + (long)s * FEAT + f4);
    float* __restrict__ ap = agg + (long)d * FEAT + f4;
    unsafeAtomicAdd(ap + 0, w * h.x);
    unsafeAtomicAdd(ap + 1, w * h.y);
    unsafeAtomicAdd(ap + 2, w * h.z);
    unsafeAtomicAdd(ap + 3, w * h.w);
}

// ---------------- finalize: out = [relu](agg + dinv^2 * H + b), float4 ------
__global__ void finalize_kernel(const float* __restrict__ agg, const float* __restrict__ H,
                                const float* __restrict__ dinv, const float* __restrict__ bias,
                                float* __restrict__ out, long N, int relu) {
    long t = (long)blockIdx.x * blockDim.x + threadIdx.x;   // over N*16 float4 chunks
    if (t >= N * (FEAT / 4)) return;
    long i = t >> 4;
    int  f4 = (int)(t & 15) * 4;
    float di = dinv[i];
    float sw = di * di;
    long  o  = i * FEAT + f4;
    const float4 a = *reinterpret_cast<const float4*>(agg + o);
    const float4 h = *reinterpret_cast<const float4*>(H + o);
    const float4 b = *reinterpret_cast<const float4*>(bias + f4);
    float4 v;
    v.x = a.x + sw * h.x + b.x;
    v.y = a.y + sw * h.y + b.y;
    v.z = a.z + sw * h.z + b.z;
    v.w = a.w + sw * h.w + b.w;
    if (relu) {
        v.x = fmaxf(v.x, 0.0f); v.y = fmaxf(v.y, 0.0f);
        v.z = fmaxf(v.z, 0.0f); v.w = fmaxf(v.w, 0.0f);
    }
    *reinterpret_cast<float4*>(out + o) = v;
}

// ---------------------------------------------------------------------------
extern "C" void kernel_launch(void* const* d_in, const int* in_sizes, int n_in,
                              void* d_out, int out_size, void* d_ws, size_t ws_size,
                              hipStream_t stream) {
    const float* x  = (const float*)d_in[0];
    const int*   ei = (const int*)  d_in[1];
    const float* W1 = (const float*)d_in[2];
    const float* b1 = (const float*)d_in[3];
    const float* W2 = (const float*)d_in[4];
    const float* b2 = (const float*)d_in[5];
    const float* W3 = (const float*)d_in[6];
    const float* b3 = (const float*)d_in[7];

    const int  D_IN = 128;
    const long N = in_sizes[0] / D_IN;      // 100000
    const long E = in_sizes[1] / 2;         // 3200000
    const int* esrc = ei;
    const int* edst = ei + E;

    float* bufA = (float*)d_out;            // layer activations (N*64)
    float* dinv = (float*)d_ws;             // N
    float* norm = dinv + N;                 // E
    float* Hbuf = norm + E;                 // N*64
    float* agg  = Hbuf + (long)N * FEAT;    // N*64

    const long NF = N * FEAT;

    // ---- degrees & symmetric normalization -------------------------------
    fill_kernel<<<1024, 256, 0, stream>>>(dinv, 1.0f, N);                    // self-loop
    deg_kernel<<<(int)((E + 255) / 256), 256, 0, stream>>>(edst, dinv, E);
    dinv_kernel<<<(int)((N + 255) / 256), 256, 0, stream>>>(dinv, N);
    norm_kernel<<<(int)((E + 255) / 256), 256, 0, stream>>>(esrc, edst, dinv, norm, E);

    const int Mtiles      = (int)(N >> 4);
    const int gemm_blocks = (Mtiles + 7) / 8;                        // 8 waves / block
    const int agg_blocks  = (int)((E * (FEAT / 4) + 255) / 256);     // 16 thr / edge
    const int fin_blocks  = (int)((N * (FEAT / 4) + 255) / 256);     // float4 chunks

    // ---- layer 1: x[128] -> 64, ReLU -------------------------------------
    gemm_wmma_kernel<<<gemm_blocks, 256, 0, stream>>>(x, W1, Hbuf, (int)N, 128);
    fill_kernel<<<2048, 256, 0, stream>>>(agg, 0.0f, NF);
    agg_kernel<<<agg_blocks, 256, 0, stream>>>(esrc, edst, norm, Hbuf, agg, E);
    finalize_kernel<<<fin_blocks, 256, 0, stream>>>(agg, Hbuf, dinv, b1, bufA, N, 1);

    // ---- layer 2: 64 -> 64, ReLU -----------------------------------------
    gemm_wmma_kernel<<<gemm_blocks, 256, 0, stream>>>(bufA, W2, Hbuf, (int)N, 64);
    fill_kernel<<<2048, 256, 0, stream>>>(agg, 0.0f, NF);
    agg_kernel<<<agg_blocks, 256, 0, stream>>>(esrc, edst, norm, Hbuf, agg, E);
    finalize_kernel<<<fin_blocks, 256, 0, stream>>>(agg, Hbuf, dinv, b2, bufA, N, 1);

    // ---- layer 3: 64 -> 64, no ReLU, writes d_out ------------------------
    gemm_wmma_kernel<<<gemm_blocks, 256, 0, stream>>>(bufA, W3, Hbuf, (int)N, 64);
    fill_kernel<<<2048, 256, 0, stream>>>(agg, 0.0f, NF);
    agg_kernel<<<agg_blocks, 256, 0, stream>>>(esrc, edst, norm, Hbuf, agg, E);
    finalize_kernel<<<fin_blocks, 256, 0, stream>>>(agg, Hbuf, dinv, b3, bufA, N, 0);
}